// Xonv2D_35794257445524
// MI455X (gfx1250) — compile-verified
//
#include <hip/hip_runtime.h>

// CDNA5 / gfx1250, wave32.
typedef __attribute__((ext_vector_type(2))) float v2f;
typedef __attribute__((ext_vector_type(8))) float v8f;

#define B_    4
#define CIN_  16
#define COUT_ 16
#define H_    128
#define W_    128
#define KK    144              // CIN*3*3 contraction length
#define WPB   8                // waves (=pixels) per block

// out[b,o,h,w] = sum_{c,kh,kw} xpad[b,c,h+kh,w+kw] * W[h,w,o,c,kh,kw] + bias[h,w,o]
// Per pixel: D(16x16) = A(16x144) * B(144x16) via 36 x V_WMMA_F32_16X16X4_F32.
//   A = per-pixel weights (contiguous 16x144 block in memory)
//   B columns 0..3 = batch samples; columns 4..15 duplicate (discarded on store)
__global__ __launch_bounds__(256) void xonv2d_wmma_f32(
    const float* __restrict__ x,     // (4,16,128,128)
    const float* __restrict__ wgt,   // (128,128,16,16,3,3): 16x144 contiguous per pixel
    const float* __restrict__ bias,  // (128,128,16)
    float* __restrict__ out)         // (4,16,128,128)
{
  __shared__ __align__(16) float patch[WPB][B_][KK];   // 18432 B

  const int lane = threadIdx.x & 31;
  const int wave = threadIdx.x >> 5;
  const int pix  = blockIdx.x * WPB + wave;            // 0..16383
  const int h    = pix >> 7;
  const int w    = pix & 127;

  // ---- Stage zero-padded input patch into LDS: patch[b][p], p = c*9+kh*3+kw ----
  float* pat = &patch[wave][0][0];
  for (int e = lane; e < B_ * KK; e += 32) {
    int b  = e / KK;
    int p  = e - b * KK;
    int c  = p / 9;
    int r  = p - c * 9;
    int kh = r / 3;
    int kw = r - kh * 3;
    int hh = h + kh - 1;
    int ww = w + kw - 1;
    float v = 0.0f;
    if (hh >= 0 && hh < H_ && ww >= 0 && ww < W_)
      v = x[((b * CIN_ + c) * H_ + hh) * W_ + ww];
    pat[e] = v;
  }
  __syncthreads();

  const float* wpix = wgt + (size_t)pix * (COUT_ * KK);  // contiguous 9216 B
  const int n    = lane & 15;   // M row for A, N column for B/D
  const int half = lane >> 4;

  // ISA 16x4 f32 A layout: lane(M=n, half) holds K = {2*half, 2*half+1} in v[0],v[1]
  // per 4-wide K step. B assumed mirrored: lane(N=n, half) holds rows {p0, p0+1}.
  v8f acc = {};
  for (int t = 0; t < KK / 4; ++t) {
    const int p0 = 4 * t + 2 * half;
    v2f a = *(const v2f*)(wpix + n * KK + p0);                 // global_load_b64
    v2f b = *(const v2f*)(pat + (n & (B_ - 1)) * KK + p0);     // ds_load_b64
    acc = __builtin_amdgcn_wmma_f32_16x16x4_f32(
        /*neg_a=*/false, a, /*neg_b=*/false, b,
        /*c_mod=*/(short)0, acc, /*reuse_a=*/false, /*reuse_b=*/false);
  }

  // D layout: VGPR r holds M = r + 8*half, N = n (lanes 0-15 / 16-31).
  if (n < B_) {
    const float* bp = bias + (size_t)pix * COUT_;
    float* op = out + (size_t)n * COUT_ * (H_ * W_) + pix;
#pragma unroll
    for (int r = 0; r < 8; ++r) {
      const int m = r + 8 * half;
      op[(size_t)m * (H_ * W_)] = acc[r] + bp[m];
    }
  }
}

extern "C" void kernel_launch(void* const* d_in, const int* in_sizes, int n_in,
                              void* d_out, int out_size, void* d_ws, size_t ws_size,
                              hipStream_t stream) {
  (void)in_sizes; (void)n_in; (void)out_size; (void)d_ws; (void)ws_size;
  const float* x    = (const float*)d_in[0];
  const float* wgt  = (const float*)d_in[1];
  const float* bias = (const float*)d_in[2];
  float* out        = (float*)d_out;

  dim3 grid((H_ * W_) / WPB);   // 2048 blocks, 8 waves (pixels) each
  dim3 block(256);
  xonv2d_wmma_f32<<<grid, block, 0, stream>>>(x, wgt, bias, out);
}